// SpeMamba_19112604467643
// MI455X (gfx1250) — compile-verified
//
#include <hip/hip_runtime.h>
#include <hip/hip_bf16.h>

#if defined(__HIP_DEVICE_COMPILE__) && !__has_builtin(__builtin_amdgcn_wmma_f32_16x16x4_f32)
#error "missing __builtin_amdgcn_wmma_f32_16x16x4_f32 on this toolchain (device pass)"
#endif

typedef __attribute__((ext_vector_type(2))) float v2f;
typedef __attribute__((ext_vector_type(8))) float v8f;

#define CCH   64
#define HH_   128
#define WW_   128
#define HWSZ  (HH_ * WW_)
#define SEQB  16          // sequences per block
#define ROWS  128         // SEQB * 8 tokens
#define NELEM (4u * CCH * HWSZ)       // 4,194,304
#define GELEM 262144                  // elems per (b, group) = 16*128*128

__device__ __forceinline__ float sigmoidf_(float v) { return 1.f / (1.f + __expf(-v)); }

// ---------------------------------------------------------------------------
// Fused Mamba kernel: 16 sequences per 256-thread block (8 waves, wave32).
// All three projections run on V_WMMA_F32_16X16X4_F32; scan stays in VGPRs.
// ---------------------------------------------------------------------------
__global__ __launch_bounds__(256)
void mamba_fused(const float* __restrict__ x,
                 const float* __restrict__ Win,  const float* __restrict__ wcv,
                 const float* __restrict__ bcv,  const float* __restrict__ Wxp,
                 const float* __restrict__ Wdt,  const float* __restrict__ bdt,
                 const float* __restrict__ Alog, const float* __restrict__ Dv,
                 const float* __restrict__ Wout, float* __restrict__ ymid)
{
    __shared__ float s_Win[256];        // W_in  (32 x 8)
    __shared__ float s_Wxp[33 * 16];    // W_xproj
    __shared__ float s_Wdt[16], s_bdt[16], s_bconv[16], s_D[16];
    __shared__ float s_wconv[64];       // (16 x 4)
    __shared__ float s_A[256];          // A = -exp(A_log), (16 x 16)
    __shared__ float s_Wout[128];       // (8 x 16)
    __shared__ float s_xf[16 * 65];     // [seq][tok*8+d], padded
    __shared__ float s_xz[ROWS * 33];   // xm: cols 0..15, z: cols 16..31
    __shared__ float s_xc[ROWS * 17];   // conv+SiLU output
    __shared__ float s_Bm[ROWS * 17];   // B projection
    __shared__ float s_Cm[ROWS * 17];   // C projection
    __shared__ float s_dt[ROWS];        // dt scalar per row
    __shared__ float s_y [ROWS * 17];   // scan output

    const int tid = threadIdx.x;
    const int n0  = blockIdx.x * SEQB;
    const int bb  = n0 >> 14;           // n0 / (128*128)
    const int rm  = n0 & 16383;
    const int hh  = rm >> 7;
    const int ww0 = rm & 127;           // 16 | 128 -> tile never crosses a row

    // ---- phase 0: stage weights + input tile into LDS -----------------
    for (int i = tid; i < 256;     i += 256) s_Win[i] = Win[i];
    for (int i = tid; i < 33 * 16; i += 256) s_Wxp[i] = Wxp[i];
    for (int i = tid; i < 256;     i += 256) s_A[i]   = -__expf(Alog[i]);
    if (tid < 16) { s_Wdt[tid] = Wdt[tid]; s_bdt[tid] = bdt[tid];
                    s_bconv[tid] = bcv[tid]; s_D[tid] = Dv[tid]; }
    if (tid < 64)  s_wconv[tid] = wcv[tid];
    if (tid < 128) s_Wout[tid]  = Wout[tid];
    #pragma unroll
    for (int j = 0; j < 4; ++j) {
        int idx = tid + j * 256;                 // 1024 = 64 ch * 16 px
        int c = idx >> 4, wl = idx & 15;
        const float* gp = x + ((size_t)(bb * CCH + c) * HH_ + hh) * WW_ + ww0 + wl;
        s_xf[wl * 65 + c] = *gp;                 // c == tok*8 + d
        __builtin_prefetch(gp + SEQB, 0, 1);     // warm next tile (global_prefetch_b8)
    }
    __syncthreads();

    const int wid    = tid >> 5, lane = tid & 31;
    const int r0     = wid * 16;                 // one 16-row M-tile per wave
    const int lrow   = r0 + (lane & 15);
    const int kh     = (lane >= 16) ? 2 : 0;     // ISA f32 A-frag K split
    const int col    = lane & 15;
    const int rstore = r0 + ((lane >= 16) ? 8 : 0); // D-frag row base

    // ---- phase 1: GEMM1  xz = xf @ W_in^T  (M=128, K=8, N=32) ---------
    {
        const int arow = (lrow >> 3) * 65 + (lrow & 7) * 8;
        #pragma unroll
        for (int nt = 0; nt < 2; ++nt) {
            v8f acc = {};
            #pragma unroll
            for (int k0 = 0; k0 < 8; k0 += 4) {
                v2f a, b;
                a.x = s_xf[arow + k0 + kh];
                a.y = s_xf[arow + k0 + kh + 1];
                int e = nt * 16 + col;
                b.x = s_Win[e * 8 + k0 + kh];
                b.y = s_Win[e * 8 + k0 + kh + 1];
                acc = __builtin_amdgcn_wmma_f32_16x16x4_f32(
                        false, a, false, b, (short)0, acc, false, false);
            }
            #pragma unroll
            for (int j = 0; j < 8; ++j)
                s_xz[(rstore + j) * 33 + nt * 16 + col] = acc[j];
        }
    }
    __syncthreads();

    // ---- phase 2: causal depthwise conv (D_CONV=4) + SiLU -------------
    // branch-free: clamp the tap row (always-valid LDS read) and select the
    // tap weight to zero for t-3+k < 0  ->  v_cndmask instead of exec-mask.
    #pragma unroll
    for (int j = 0; j < 8; ++j) {
        int idx = tid + j * 256;                 // 2048 = 16*8*16
        int i = idx & 15, t = (idx >> 4) & 7, sq = idx >> 7;
        float acc = s_bconv[i];
        #pragma unroll
        for (int k = 0; k < 4; ++k) {
            int tt  = t - 3 + k;
            int ttc = (tt < 0) ? 0 : tt;
            float w = (tt < 0) ? 0.f : s_wconv[i * 4 + k];
            acc += s_xz[(sq * 8 + ttc) * 33 + i] * w;
        }
        s_xc[(sq * 8 + t) * 17 + i] = acc * sigmoidf_(acc);
    }
    __syncthreads();

    // ---- phase 3: dt row-scalar + GEMM2 (B,C)  (M=128, K=16, N=2x16) --
    if (tid < ROWS) {
        float a = 0.f;
        #pragma unroll
        for (int i = 0; i < 16; ++i) a += s_xc[tid * 17 + i] * s_Wxp[i]; // W_xproj row 0
        s_dt[tid] = a;
    }
    #pragma unroll
    for (int nt = 0; nt < 2; ++nt) {
        v8f acc = {};
        #pragma unroll
        for (int k0 = 0; k0 < 16; k0 += 4) {
            v2f a, b;
            a.x = s_xc[lrow * 17 + k0 + kh];
            a.y = s_xc[lrow * 17 + k0 + kh + 1];
            int e = 1 + nt * 16 + col;           // rows 1..16 = B, 17..32 = C
            b.x = s_Wxp[e * 16 + k0 + kh];
            b.y = s_Wxp[e * 16 + k0 + kh + 1];
            acc = __builtin_amdgcn_wmma_f32_16x16x4_f32(
                    false, a, false, b, (short)0, acc, false, false);
        }
        float* dst = nt ? s_Cm : s_Bm;
        #pragma unroll
        for (int j = 0; j < 8; ++j)
            dst[(rstore + j) * 17 + col] = acc[j];
    }
    __syncthreads();

    // ---- phase 4: selective scan, 1 lane per (seq, d_inner) -----------
    {
        const int sq = tid >> 4, ii = tid & 15;
        float hst[16], Arow[16];
        #pragma unroll
        for (int s = 0; s < 16; ++s) { hst[s] = 0.f; Arow[s] = s_A[ii * 16 + s]; }
        const float wdt_i = s_Wdt[ii], bdt_i = s_bdt[ii], d_i = s_D[ii];
        #pragma unroll
        for (int t = 0; t < 8; ++t) {
            const int rb = sq * 8 + t;
            float pre   = s_dt[rb] * wdt_i + bdt_i;
            float delta = (pre > 20.f) ? pre : log1pf(__expf(pre)); // softplus
            float xci   = s_xc[rb * 17 + ii];
            float du    = delta * xci;
            float y     = 0.f;
            #pragma unroll
            for (int s = 0; s < 16; ++s) {
                float dA = __expf(delta * Arow[s]);
                hst[s] = dA * hst[s] + du * s_Bm[rb * 17 + s];
                y += hst[s] * s_Cm[rb * 17 + s];
            }
            y += d_i * xci;
            float zv = s_xz[rb * 33 + 16 + ii];
            y *= zv * sigmoidf_(zv);             // * silu(z)
            s_y[rb * 17 + ii] = y;
        }
    }
    __syncthreads();

    // ---- phase 5: GEMM3  out = y @ W_out^T (N=8, padded to 16) --------
    {
        v8f acc = {};
        #pragma unroll
        for (int k0 = 0; k0 < 16; k0 += 4) {
            v2f a, b;
            a.x = s_y[lrow * 17 + k0 + kh];
            a.y = s_y[lrow * 17 + k0 + kh + 1];
            b.x = (col < 8) ? s_Wout[col * 16 + k0 + kh]     : 0.f;
            b.y = (col < 8) ? s_Wout[col * 16 + k0 + kh + 1] : 0.f;
            acc = __builtin_amdgcn_wmma_f32_16x16x4_f32(
                    false, a, false, b, (short)0, acc, false, false);
        }
        if (col < 8) {
            #pragma unroll
            for (int j = 0; j < 8; ++j) {
                int row = rstore + j;
                int sq = row >> 3, t = row & 7;
                int c = t * 8 + col;             // back to channel dim
                ymid[((size_t)(bb * CCH + c) * HH_ + hh) * WW_ + ww0 + sq] = acc[j];
            }
        }
    }
}

// ---------------------------------------------------------------------------
// GroupNorm stage 1: per-(b,group) slice partial sums (deterministic).
// grid = 16 groups * 64 slices; each block reduces 4096 contiguous floats.
// ---------------------------------------------------------------------------
__global__ __launch_bounds__(256)
void gn_partial(const float* __restrict__ y, float* __restrict__ partial)
{
    __shared__ float rs[256], rq[256];
    const int bg = blockIdx.x >> 6;
    const int sl = blockIdx.x & 63;
    const float4* p = (const float4*)(y + (size_t)bg * GELEM + (size_t)sl * 4096);
    float s = 0.f, q = 0.f;
    #pragma unroll
    for (int j = 0; j < 4; ++j) {
        float4 v = p[threadIdx.x + j * 256];
        s += v.x + v.y + v.z + v.w;
        q += v.x * v.x + v.y * v.y + v.z * v.z + v.w * v.w;
    }
    rs[threadIdx.x] = s; rq[threadIdx.x] = q;
    __syncthreads();
    for (int off = 128; off > 0; off >>= 1) {
        if (threadIdx.x < off) {
            rs[threadIdx.x] += rs[threadIdx.x + off];
            rq[threadIdx.x] += rq[threadIdx.x + off];
        }
        __syncthreads();
    }
    if (threadIdx.x == 0) {
        partial[blockIdx.x * 2]     = rs[0];
        partial[blockIdx.x * 2 + 1] = rq[0];
    }
}

// GroupNorm stage 2: fold 64 slice partials per group into sum/sumsq.
__global__ void gn_finalize(const float* __restrict__ partial, float* __restrict__ stats)
{
    int t = threadIdx.x;
    if (t < 32) {
        int bg = t >> 1, w = t & 1;
        float s = 0.f;
        for (int j = 0; j < 64; ++j) s += partial[(bg * 64 + j) * 2 + w];
        stats[bg * 2 + w] = s;
    }
}

// GroupNorm stage 3: normalize + gamma/beta + SiLU + residual (float4 path).
__global__ __launch_bounds__(256)
void gn_apply(const float* __restrict__ x, const float* __restrict__ y,
              const float* __restrict__ stats, const float* __restrict__ gamma,
              const float* __restrict__ beta, float* __restrict__ out)
{
    size_t i4 = (size_t)blockIdx.x * 256 + threadIdx.x;   // float4 index
    size_t e0 = i4 * 4;
    int c  = (int)((e0 >> 14) & 63);
    int bg = (int)(e0 >> 18);                             // e0 / 262144
    const float inv  = 1.f / (float)GELEM;
    float mean = stats[bg * 2] * inv;
    float var  = stats[bg * 2 + 1] * inv - mean * mean;
    float rstd = rsqrtf(var + 1e-5f);
    float ga = gamma[c] * rstd;
    float be = beta[c] - mean * ga;
    float4 yv = ((const float4*)y)[i4];
    float4 xv = ((const float4*)x)[i4];
    float4 o;
    { float yn = yv.x * ga + be; o.x = xv.x + yn * sigmoidf_(yn); }
    { float yn = yv.y * ga + be; o.y = xv.y + yn * sigmoidf_(yn); }
    { float yn = yv.z * ga + be; o.z = xv.z + yn * sigmoidf_(yn); }
    { float yn = yv.w * ga + be; o.w = xv.w + yn * sigmoidf_(yn); }
    ((float4*)out)[i4] = o;
}

// ---------------------------------------------------------------------------
extern "C" void kernel_launch(void* const* d_in, const int* in_sizes, int n_in,
                              void* d_out, int out_size, void* d_ws, size_t ws_size,
                              hipStream_t stream)
{
    (void)in_sizes; (void)n_in; (void)out_size; (void)ws_size;
    const float* x    = (const float*)d_in[0];
    const float* Win  = (const float*)d_in[1];
    const float* wcv  = (const float*)d_in[2];
    const float* bcv  = (const float*)d_in[3];
    const float* Wxp  = (const float*)d_in[4];
    const float* Wdt  = (const float*)d_in[5];
    const float* bdt  = (const float*)d_in[6];
    const float* Alog = (const float*)d_in[7];
    const float* Dv   = (const float*)d_in[8];
    const float* Wout = (const float*)d_in[9];
    const float* gam  = (const float*)d_in[10];
    const float* bet  = (const float*)d_in[11];
    float* out     = (float*)d_out;
    float* ymid    = (float*)d_ws;              // 4,194,304 floats (16 MB)
    float* partial = ymid + NELEM;              // 2048 floats
    float* stats   = partial + 2048;            // 32 floats

    mamba_fused<<<NELEM / (CCH * SEQB) , 256, 0, stream>>>(   // 4096 blocks
        x, Win, wcv, bcv, Wxp, Wdt, bdt, Alog, Dv, Wout, ymid);
    gn_partial <<<1024, 256, 0, stream>>>(ymid, partial);
    gn_finalize<<<1,    64,  0, stream>>>(partial, stats);
    gn_apply   <<<NELEM / 1024, 256, 0, stream>>>(x, ymid, stats, gam, bet, out);
}